// DigitCaps_62569083568319
// MI455X (gfx1250) — compile-verified
//
#include <hip/hip_runtime.h>

// DigitCaps dynamic routing, fully fused recompute strategy for MI455X (gfx1250).
// f32 WMMA 16x16x4: A = W[r,j,:,:] (16x8, two K=4 chunks), B = x[btile, r, :] (8x16),
// C/D = u_hat tile [d=16 x b=16] in v8f accumulators.

typedef __attribute__((ext_vector_type(2))) float v2f;
typedef __attribute__((ext_vector_type(8))) float v8f;

#define R_TOT   1152
#define J_CAPS  10
#define C_IN    8
#define D_DIM   16
#define BATCH   128

#define RPW     9                 // routes per wave
#define WAVES   4                 // waves per block (128 threads, wave32)
#define RPB     (RPW * WAVES)     // 36 routes per block
#define RBLK    (R_TOT / RPB)     // 32 route-blocks
#define BTILES  (BATCH / 16)      // 8 batch tiles

// ---- one 16x16 u_hat tile for (r, j, batch-tile) via two chained f32 WMMAs ----
__device__ __forceinline__ v8f uhat_tile(const float* __restrict__ W,
                                         int r, int j, int l16, int half,
                                         v2f bx0, v2f bx1)
{
    const float* wp = W + (size_t)r * (J_CAPS * D_DIM * C_IN)
                        + j * (D_DIM * C_IN) + l16 * C_IN;
    // A-matrix 16x4 f32 layout: VGPR0 = K(2*half), VGPR1 = K(2*half+1)
    v2f a0 = *(const v2f*)(wp + 2 * half);       // c = {0,1} or {2,3}
    v2f a1 = *(const v2f*)(wp + 4 + 2 * half);   // c = {4,5} or {6,7}
    v8f acc = {};
    acc = __builtin_amdgcn_wmma_f32_16x16x4_f32(false, a0, false, bx0,
                                                (short)0, acc, false, false);
    acc = __builtin_amdgcn_wmma_f32_16x16x4_f32(false, a1, false, bx1,
                                                (short)0, acc, false, false);
    return acc;
}

// ---- zero-init b[r,j] and s[b,j,d] scratch ----
__global__ void k_zero(float* __restrict__ p, int n)
{
    int t = blockIdx.x * blockDim.x + threadIdx.x;
    if (t < n) p[t] = 0.0f;
}

// ---- s[b,j,d] += sum_r softmax_j(b[r,:]) * u_hat[b,r,j,d] ----
__global__ __launch_bounds__(128) void k_accum_s(
    const float* __restrict__ x, const float* __restrict__ W,
    const float* __restrict__ barr, float* __restrict__ s)
{
    __shared__ float s_lds[J_CAPS * 16 * 16];   // [j][d][b16] = 2560 floats
    const int tid = threadIdx.x;
    for (int i = tid; i < J_CAPS * 256; i += 128) s_lds[i] = 0.0f;
    __syncthreads();

    const int wave = tid >> 5;
    const int lane = tid & 31;
    const int half = lane >> 4;
    const int l16  = lane & 15;
    const int b0   = blockIdx.y * 16;
    const int bb   = b0 + l16;

    v8f sj[J_CAPS];
    #pragma unroll
    for (int j = 0; j < J_CAPS; ++j) sj[j] = {};

    const int rbase = blockIdx.x * RPB + wave * RPW;
    for (int i = 0; i < RPW; ++i) {
        const int r = rbase + i;

        // softmax over j of b[r,:] (uniform across lanes; 10 values)
        float brow[J_CAPS];
        float m = -3.4e38f;
        #pragma unroll
        for (int j = 0; j < J_CAPS; ++j) {
            brow[j] = barr[r * J_CAPS + j];
            m = fmaxf(m, brow[j]);
        }
        float sum = 0.0f;
        #pragma unroll
        for (int j = 0; j < J_CAPS; ++j) { brow[j] = __expf(brow[j] - m); sum += brow[j]; }
        const float inv = 1.0f / sum;

        // B operand: x[bb, r, c], 4x16 f32 per K-group
        const float* xp = x + (size_t)bb * (R_TOT * C_IN) + (size_t)r * C_IN;
        v2f bx0 = *(const v2f*)(xp + 2 * half);
        v2f bx1 = *(const v2f*)(xp + 4 + 2 * half);

        #pragma unroll
        for (int j = 0; j < J_CAPS; ++j) {
            v8f acc = uhat_tile(W, r, j, l16, half, bx0, bx1);
            const float cj = brow[j] * inv;
            #pragma unroll
            for (int v = 0; v < 8; ++v) sj[j][v] += cj * acc[v];
        }
    }

    // wave partials -> LDS (ds_add_f32)
    #pragma unroll
    for (int j = 0; j < J_CAPS; ++j)
        #pragma unroll
        for (int v = 0; v < 8; ++v)
            atomicAdd(&s_lds[j * 256 + (v + 8 * half) * 16 + l16], sj[j][v]);
    __syncthreads();

    // block partial -> global s (global_atomic_add_f32)
    for (int i = tid; i < J_CAPS * 256; i += 128) {
        const int j = i >> 8, rem = i & 255, d = rem >> 4, bl = rem & 15;
        atomicAdd(&s[(size_t)(b0 + bl) * (J_CAPS * D_DIM) + j * D_DIM + d], s_lds[i]);
    }
}

// ---- v = squash(s); re-zero s; last iter writes to d_out ----
__global__ void k_squash(float* __restrict__ s, float* __restrict__ vout)
{
    const int t = blockIdx.x * blockDim.x + threadIdx.x;
    if (t >= BATCH * J_CAPS) return;
    float vals[D_DIM];
    float sqn = 0.0f;
    float* sp = s + (size_t)t * D_DIM;
    #pragma unroll
    for (int d = 0; d < D_DIM; ++d) { vals[d] = sp[d]; sqn += vals[d] * vals[d]; }
    const float scale = sqn / ((1.0f + sqn) * sqrtf(sqn));
    float* vp = vout + (size_t)t * D_DIM;
    #pragma unroll
    for (int d = 0; d < D_DIM; ++d) { vp[d] = vals[d] * scale; sp[d] = 0.0f; }
}

// ---- b[r,j] += (1/B) * sum_b sum_d u_hat[b,r,j,d] * v[b,j,d] ----
__global__ __launch_bounds__(128) void k_update_b(
    const float* __restrict__ x, const float* __restrict__ W,
    const float* __restrict__ vbuf, float* __restrict__ barr)
{
    __shared__ float v_lds[J_CAPS * 256];       // [j][d][b16]
    const int tid = threadIdx.x;
    const int b0 = blockIdx.y * 16;
    for (int i = tid; i < J_CAPS * 256; i += 128) {
        const int j = i >> 8, rem = i & 255, d = rem >> 4, bl = rem & 15;
        v_lds[i] = vbuf[(size_t)(b0 + bl) * (J_CAPS * D_DIM) + j * D_DIM + d];
    }
    __syncthreads();

    const int wave = tid >> 5;
    const int lane = tid & 31;
    const int half = lane >> 4;
    const int l16  = lane & 15;
    const int bb   = b0 + l16;

    const int rbase = blockIdx.x * RPB + wave * RPW;
    for (int i = 0; i < RPW; ++i) {
        const int r = rbase + i;
        const float* xp = x + (size_t)bb * (R_TOT * C_IN) + (size_t)r * C_IN;
        v2f bx0 = *(const v2f*)(xp + 2 * half);
        v2f bx1 = *(const v2f*)(xp + 4 + 2 * half);

        #pragma unroll
        for (int j = 0; j < J_CAPS; ++j) {
            v8f acc = uhat_tile(W, r, j, l16, half, bx0, bx1);
            float p = 0.0f;
            #pragma unroll
            for (int v = 0; v < 8; ++v)
                p += acc[v] * v_lds[j * 256 + (v + 8 * half) * 16 + l16];
            // wave32 reduction over 16 batch lanes x 2 d-halves
            #pragma unroll
            for (int off = 16; off > 0; off >>= 1)
                p += __shfl_xor(p, off, 32);
            if (lane == 0)
                atomicAdd(&barr[r * J_CAPS + j], p * (1.0f / (float)BATCH));
        }
    }
}

extern "C" void kernel_launch(void* const* d_in, const int* in_sizes, int n_in,
                              void* d_out, int out_size, void* d_ws, size_t ws_size,
                              hipStream_t stream)
{
    const float* x = (const float*)d_in[0];   // [128, 1152, 8]
    const float* W = (const float*)d_in[1];   // [1, 1152, 10, 16, 8]

    float* barr = (float*)d_ws;               // [1152,10]   = 11520 floats
    float* s    = barr + R_TOT * J_CAPS;      // [128,10,16] = 20480 floats
    float* vbuf = s + BATCH * J_CAPS * D_DIM; // [128,10,16] = 20480 floats

    const int nz = R_TOT * J_CAPS + BATCH * J_CAPS * D_DIM;  // barr + s contiguous
    k_zero<<<(nz + 255) / 256, 256, 0, stream>>>(barr, nz);

    dim3 grid(RBLK, BTILES);
    for (int it = 0; it < 3; ++it) {
        k_accum_s<<<grid, 128, 0, stream>>>(x, W, barr, s);
        float* vdst = (it == 2) ? (float*)d_out : vbuf;
        k_squash<<<(BATCH * J_CAPS + 255) / 256, 256, 0, stream>>>(s, vdst);
        if (it < 2)
            k_update_b<<<grid, 128, 0, stream>>>(x, W, vbuf, barr);
    }
}